// MyLayer_12120397709448
// MI455X (gfx1250) — compile-verified
//
#include <hip/hip_runtime.h>
#include <math.h>

typedef __attribute__((ext_vector_type(2))) float v2f;
typedef __attribute__((ext_vector_type(8))) float v8f;

#define NEG_SLOPE 0.01f

__device__ __forceinline__ float lrelu(float v) {
  return v > 0.0f ? v : NEG_SLOPE * v;
}

// Device-scope fire-and-forget float add -> global_atomic_add_f32 (no CAS loop).
__device__ __forceinline__ void atomic_add_agent(float* p, float v) {
  __hip_atomic_fetch_add(p, v, __ATOMIC_RELAXED, __HIP_MEMORY_SCOPE_AGENT);
}
// Workgroup-scope LDS float add -> ds_add_f32.
__device__ __forceinline__ void atomic_add_wg(float* p, float v) {
  __hip_atomic_fetch_add(p, v, __ATOMIC_RELAXED, __HIP_MEMORY_SCOPE_WORKGROUP);
}

// ---------------------------------------------------------------------------
// C[M,NC] = A[M,K] @ W[NC,K]^T  via V_WMMA_F32_16X16X4_F32, one wave per
// 16x16 output tile. Requires M%16==0, K%4==0, NC%16==0 (all hold here:
// M=50000, K in {20,32}, NC in {32,64}).
//
// ISA 7.12.2 layouts (wave32):
//   A 16x4 f32 : lanes 0-15 hold row M=l, VGPR0=K0,VGPR1=K1;
//                lanes 16-31 hold row M=l, VGPR0=K2,VGPR1=K3.
//   B 4x16 f32 : mirrored (lane-within-half = N, half selects K pair).
//   C/D 16x16  : VGPR r -> M=r (lanes 0-15) / M=8+r (lanes 16-31), N=l.
// ---------------------------------------------------------------------------
__global__ void wmma_gemm_nt(const float* __restrict__ A,
                             const float* __restrict__ W,
                             float* __restrict__ C,
                             int M, int K, int NC) {
  const int wave = (int)((blockIdx.x * blockDim.x + threadIdx.x) >> 5);
  const int lane = (int)(threadIdx.x & 31);
  const int jt = NC >> 4;
  const int tiles = (M >> 4) * jt;
  if (wave >= tiles) return;            // wave-uniform: EXEC stays all-ones
  const int m0 = (wave / jt) << 4;
  const int j0 = (wave % jt) << 4;
  const int half = lane >> 4;           // selects K pair {0,1} vs {2,3}
  const int l = lane & 15;              // A row / B column within tile
  const float* arow = A + (size_t)(m0 + l) * K + half * 2;
  const float* wrow = W + (size_t)(j0 + l) * K + half * 2;   // W row j = B col j
  v8f c = {};
  for (int k = 0; k < K; k += 4) {
    v2f a, b;
    a.x = arow[k];     a.y = arow[k + 1];
    b.x = wrow[k];     b.y = wrow[k + 1];
    c = __builtin_amdgcn_wmma_f32_16x16x4_f32(false, a, false, b,
                                              (short)0, c, false, false);
  }
  float* crow = C + (size_t)(m0 + half * 8) * NC + j0 + l;
#pragma unroll
  for (int r = 0; r < 8; ++r) crow[(size_t)r * NC] = c[r];
}

// ---------------------------------------------------------------------------
// Zero a contiguous float region (accumulators must be re-zeroed every call).
// ---------------------------------------------------------------------------
__global__ void zero_f32(float* __restrict__ p, size_t n) {
  size_t i = (size_t)blockIdx.x * blockDim.x + threadIdx.x;
  const size_t stride = (size_t)gridDim.x * blockDim.x;
  for (; i < n; i += stride) p[i] = 0.0f;
}

// ---------------------------------------------------------------------------
// Layer-1 edge pass: one wave per edge, lane = feature (32 features).
// msg = lrelu(w_e * h1[col] + b1); scatter-add into acc[row]; count rows.
// ---------------------------------------------------------------------------
__global__ void edge_msg32(const int* __restrict__ eidx,
                           const float* __restrict__ wgt,
                           const float* __restrict__ h,
                           const float* __restrict__ bias,
                           float* __restrict__ acc,
                           float* __restrict__ cnt, int E) {
  const int e = (int)((blockIdx.x * blockDim.x + threadIdx.x) >> 5);
  const int lane = (int)(threadIdx.x & 31);
  if (e >= E) return;
  const int row = eidx[e];
  const int col = eidx[E + e];
  const float w = wgt[e];
  float v = fmaf(w, h[(size_t)col * 32 + lane], bias[lane]);
  v = lrelu(v);
  atomic_add_agent(&acc[(size_t)row * 32 + lane], v);
  if (lane == 0) atomic_add_agent(&cnt[row], 1.0f);
}

// Layer-2 edge pass: 64 features, 2 per lane.
__global__ void edge_msg64(const int* __restrict__ eidx,
                           const float* __restrict__ wgt,
                           const float* __restrict__ h,
                           const float* __restrict__ bias,
                           float* __restrict__ acc, int E) {
  const int e = (int)((blockIdx.x * blockDim.x + threadIdx.x) >> 5);
  const int lane = (int)(threadIdx.x & 31);
  if (e >= E) return;
  const int row = eidx[e];
  const int col = eidx[E + e];
  const float w = wgt[e];
#pragma unroll
  for (int t = 0; t < 2; ++t) {
    const int j = lane + t * 32;
    float v = fmaf(w, h[(size_t)col * 64 + j], bias[j]);
    v = lrelu(v);
    atomic_add_agent(&acc[(size_t)row * 64 + j], v);
  }
}

// ---------------------------------------------------------------------------
// Layer-1 node combine (in place into acc): out1 = acc/max(cnt,1)+lrelu(h+b).
// ---------------------------------------------------------------------------
__global__ void node_out32(float* __restrict__ acc,
                           const float* __restrict__ cnt,
                           const float* __restrict__ h,
                           const float* __restrict__ bias, int N) {
  const int n = (int)((blockIdx.x * blockDim.x + threadIdx.x) >> 5);
  const int lane = (int)(threadIdx.x & 31);
  if (n >= N) return;
  const float d = fmaxf(cnt[n], 1.0f);
  const size_t i = (size_t)n * 32 + lane;
  acc[i] = acc[i] / d + lrelu(h[i] + bias[lane]);
}

// ---------------------------------------------------------------------------
// Layer-2 node combine fused with global mean-pool numerator. Grid-stride
// over nodes; per-block LDS reduction (ds_add_f32) -> 64 global atomics/block.
// out2 is never materialized.
// ---------------------------------------------------------------------------
__global__ void node_out64_pool(const float* __restrict__ acc,
                                const float* __restrict__ cnt,
                                const float* __restrict__ h,
                                const float* __restrict__ bias,
                                float* __restrict__ pooled, int N) {
  __shared__ float sh[64];
  if (threadIdx.x < 64) sh[threadIdx.x] = 0.0f;
  __syncthreads();
  const int lane = (int)(threadIdx.x & 31);
  const int wib = (int)(threadIdx.x >> 5);
  const int wpb = (int)(blockDim.x >> 5);
  const int stride = (int)gridDim.x * wpb;
  float s0 = 0.0f, s1 = 0.0f;
  for (int n = (int)blockIdx.x * wpb + wib; n < N; n += stride) {
    const float d = fmaxf(cnt[n], 1.0f);
    const size_t base = (size_t)n * 64;
    s0 += acc[base + lane]      / d + lrelu(h[base + lane]      + bias[lane]);
    s1 += acc[base + lane + 32] / d + lrelu(h[base + lane + 32] + bias[lane + 32]);
  }
  atomic_add_wg(&sh[lane], s0);
  atomic_add_wg(&sh[lane + 32], s1);
  __syncthreads();
  if (threadIdx.x < 64) atomic_add_agent(&pooled[threadIdx.x], sh[threadIdx.x]);
}

// ---------------------------------------------------------------------------
// pooled/N @ W7^T + b7, then 2-class log-softmax -> d_out[0..1].
// ---------------------------------------------------------------------------
__global__ void finalize(const float* __restrict__ pooled,
                         const float* __restrict__ W7,
                         const float* __restrict__ b7,
                         float invN, float* __restrict__ out) {
  if (threadIdx.x == 0 && blockIdx.x == 0) {
    float l0 = b7[0], l1 = b7[1];
    for (int j = 0; j < 64; ++j) {
      const float p = pooled[j] * invN;
      l0 = fmaf(p, W7[j], l0);
      l1 = fmaf(p, W7[64 + j], l1);
    }
    const float m = fmaxf(l0, l1);
    const float lse = m + logf(expf(l0 - m) + expf(l1 - m));
    out[0] = l0 - lse;
    out[1] = l1 - lse;
  }
}

extern "C" void kernel_launch(void* const* d_in, const int* in_sizes, int n_in,
                              void* d_out, int out_size, void* d_ws, size_t ws_size,
                              hipStream_t stream) {
  (void)n_in; (void)out_size; (void)ws_size;
  const float* x    = (const float*)d_in[0];   // [N,20]
  const int*   edge = (const int*)  d_in[1];   // [2,E] (row = [0,E), col = [E,2E))
  const float* wgt  = (const float*)d_in[2];   // [E]
  const float* W1   = (const float*)d_in[3];   // [32,20]
  const float* b1   = (const float*)d_in[4];   // [32]
  const float* W3   = (const float*)d_in[5];   // [64,32]
  const float* b3   = (const float*)d_in[6];   // [64]
  const float* W7   = (const float*)d_in[7];   // [2,64]
  const float* b7   = (const float*)d_in[8];   // [2]
  float* out = (float*)d_out;

  const int N = in_sizes[0] / 20;
  const int E = in_sizes[2];

  // Workspace: h1[N*32] | h2[N*64] | acc1[N*32] | acc2[N*64] | cnt[N] | pooled[64]
  float* ws     = (float*)d_ws;
  float* h1     = ws;
  float* h2     = h1   + (size_t)N * 32;
  float* acc1   = h2   + (size_t)N * 64;
  float* acc2   = acc1 + (size_t)N * 32;
  float* cnt    = acc2 + (size_t)N * 64;
  float* pooled = cnt  + (size_t)N;

  // 1) zero acc1..pooled (contiguous: N*32 + N*64 + N + 64 floats)
  zero_f32<<<2048, 256, 0, stream>>>(acc1, (size_t)N * 97 + 64);

  // 2) h1 = x @ W1^T  (fp32 WMMA, K=20)
  {
    const int tiles = (N / 16) * (32 / 16);
    wmma_gemm_nt<<<(tiles + 7) / 8, 256, 0, stream>>>(x, W1, h1, N, 20, 32);
  }

  // 3) layer-1 edge scatter (+ per-row counts, reused by layer 2)
  edge_msg32<<<(E + 7) / 8, 256, 0, stream>>>(edge, wgt, h1, b1, acc1, cnt, E);

  // 4) out1 = acc1/max(cnt,1) + lrelu(h1+b1), in place -> acc1
  node_out32<<<(N + 7) / 8, 256, 0, stream>>>(acc1, cnt, h1, b1, N);

  // 5) h2 = out1 @ W3^T  (fp32 WMMA, K=32)
  {
    const int tiles = (N / 16) * (64 / 16);
    wmma_gemm_nt<<<(tiles + 7) / 8, 256, 0, stream>>>(acc1, W3, h2, N, 32, 64);
  }

  // 6) layer-2 edge scatter
  edge_msg64<<<(E + 7) / 8, 256, 0, stream>>>(edge, wgt, h2, b3, acc2, E);

  // 7) layer-2 node combine fused with mean-pool sum
  node_out64_pool<<<512, 256, 0, stream>>>(acc2, cnt, h2, b3, pooled, N);

  // 8) classifier + log_softmax
  finalize<<<1, 32, 0, stream>>>(pooled, W7, b7, 1.0f / (float)N, out);
}